// GCN_Mult_3770981285985
// MI455X (gfx1250) — compile-verified
//
#include <hip/hip_runtime.h>

typedef float v2f __attribute__((ext_vector_type(2)));
typedef float v8f __attribute__((ext_vector_type(8)));

#define NFEAT 256
#define NHID  64

// ---------------- degree / norm kernels ----------------

__global__ void k_deg_init(float* deg, int n) {
    int i = blockIdx.x * blockDim.x + threadIdx.x;
    if (i < n) deg[i] = 1.0f;   // self-loop contribution
}

__global__ void k_deg_count(const int* __restrict__ dst, float* deg, int n_edges) {
    int e = blockIdx.x * blockDim.x + threadIdx.x;
    if (e < n_edges) atomicAdd(&deg[dst[e]], 1.0f);
}

__global__ void k_dinv(float* deg, int n) {
    int i = blockIdx.x * blockDim.x + threadIdx.x;
    if (i < n) {
        float d = deg[i];
        deg[i] = (d > 0.0f) ? rsqrtf(d) : 0.0f;  // in-place: deg -> deg^{-1/2}
    }
}

// ---------------- layer-1 dense transform: h = x @ W1 (WMMA fp32) ----------------
// block = 256 threads = 8 waves; each wave computes a 16x64 tile of h.
// W1 (256x64 fp32 = 64KB) staged in LDS in K-pair-interleaved, bank-rotated
// layout: element W1[k][n] lives at  p*128 + ((2n + (p&1)*32) & 127) + (k&1),
// with p = k/2.  A lane's B fragment {W[kb][n], W[kb+1][n]} is then one
// aligned 8-byte word -> single conflict-free ds_load_b64 per fragment.

__global__ __launch_bounds__(256) void k_gemm1_wmma(const float* __restrict__ x,
                                                    const float* __restrict__ W1,
                                                    float* __restrict__ h,
                                                    int n_nodes) {
    __shared__ float ldsW[NFEAT * NHID];   // 64 KB (128 pair-rows x 128 words)
    int tid = threadIdx.x;

    // cooperative staging: float4 global loads, scattered b32 LDS stores
    for (int idx = tid * 4; idx < NFEAT * NHID; idx += 256 * 4) {
        float4 w = *(const float4*)(W1 + idx);
        int k  = idx >> 6;          // K row (same for all 4 elements)
        int n0 = idx & 63;          // first N column
        int p  = k >> 1;
        int lo = k & 1;
        int rot = (p & 1) << 5;
        float vals[4] = {w.x, w.y, w.z, w.w};
        #pragma unroll
        for (int j = 0; j < 4; ++j) {
            int wpos = (((n0 + j) << 1) + rot) & 127;
            ldsW[p * 128 + wpos + lo] = vals[j];
        }
    }
    __syncthreads();

    int wave = tid >> 5;
    int lane = tid & 31;
    int tile = blockIdx.x * 8 + wave;   // 16-row tile index (wave-uniform)
    int m0 = tile * 16;
    if (m0 >= n_nodes) return;          // uniform per wave: EXEC stays all-1s

    int half = lane >> 4;               // 0: K=k..k+1, 1: K=k+2..k+3
    int mrow = m0 + (lane & 15);        // A row for this lane
    int kk   = half << 1;
    int ncol = lane & 15;               // B/D column within a 16-wide N tile
    int rot  = half << 5;               // pair parity == half for k%4==0 steps

    // loop-invariant word positions of the 4 B fragments for this lane
    int wp0 = (((ncol +  0) << 1) + rot) & 127;
    int wp1 = (((ncol + 16) << 1) + rot) & 127;
    int wp2 = (((ncol + 32) << 1) + rot) & 127;
    int wp3 = (((ncol + 48) << 1) + rot) & 127;

    v8f acc0 = {}, acc1 = {}, acc2 = {}, acc3 = {};

    const float* xrow = x + (size_t)mrow * NFEAT;
    for (int k = 0; k < NFEAT; k += 4) {
        // A fragment: lane holds {x[m, k+kk], x[m, k+kk+1]}
        float2 av = *(const float2*)(xrow + k + kk);
        v2f a; a.x = av.x; a.y = av.y;

        // B fragments: pair-row p = k/2 + half, single b64 load each
        const float* rowp = ldsW + ((k >> 1) + half) * 128;
        v2f b0 = *(const v2f*)(rowp + wp0);
        v2f b1 = *(const v2f*)(rowp + wp1);
        v2f b2 = *(const v2f*)(rowp + wp2);
        v2f b3 = *(const v2f*)(rowp + wp3);

        acc0 = __builtin_amdgcn_wmma_f32_16x16x4_f32(false, a, false, b0, (short)0, acc0, false, false);
        acc1 = __builtin_amdgcn_wmma_f32_16x16x4_f32(false, a, false, b1, (short)0, acc1, false, false);
        acc2 = __builtin_amdgcn_wmma_f32_16x16x4_f32(false, a, false, b2, (short)0, acc2, false, false);
        acc3 = __builtin_amdgcn_wmma_f32_16x16x4_f32(false, a, false, b3, (short)0, acc3, false, false);
    }

    // D layout: VGPR r -> lanes 0-15: M=r, N=lane; lanes 16-31: M=8+r, N=lane-16
    int rbase = m0 + (half << 3);
    for (int r = 0; r < 8; ++r) {
        float* hr = h + (size_t)(rbase + r) * NHID;
        hr[ncol +  0] = acc0[r];
        hr[ncol + 16] = acc1[r];
        hr[ncol + 32] = acc2[r];
        hr[ncol + 48] = acc3[r];
    }
}

// ---------------- layer-1 aggregation ----------------

// out1[i][f] = h[i][f] * dinv[i]^2   (self-loop term; also fully initializes out1)
__global__ void k_selfloop1(const float* __restrict__ h, const float* __restrict__ dinv,
                            float* __restrict__ out1, int n_nodes) {
    int t = blockIdx.x * blockDim.x + threadIdx.x;
    int i = t >> 6, f = t & 63;
    if (i >= n_nodes) return;
    float di = dinv[i];
    out1[(size_t)i * NHID + f] = h[(size_t)i * NHID + f] * di * di;
}

// out1[dst][f] += h[src][f] * dinv[src]*dinv[dst]
__global__ void k_edge_agg1(const int* __restrict__ src, const int* __restrict__ dst,
                            const float* __restrict__ dinv, const float* __restrict__ h,
                            float* __restrict__ out1, int n_edges) {
    int t = blockIdx.x * blockDim.x + threadIdx.x;
    int e = t >> 6, f = t & 63;
    if (e >= n_edges) return;
    int s = src[e], d = dst[e];
    float norm = dinv[s] * dinv[d];
    atomicAdd(&out1[(size_t)d * NHID + f], h[(size_t)s * NHID + f] * norm);
}

// h1[i][f] = relu(out1[i][f] + b1[f])   (written into the h buffer, reused)
__global__ void k_relu_bias(const float* __restrict__ out1, const float* __restrict__ b1,
                            float* __restrict__ h1, int n_nodes) {
    int t = blockIdx.x * blockDim.x + threadIdx.x;
    int i = t >> 6, f = t & 63;
    if (i >= n_nodes) return;
    float v = out1[(size_t)i * NHID + f] + b1[f];
    h1[(size_t)i * NHID + f] = fmaxf(v, 0.0f);
}

// ---------------- layer 2: z = h1 @ W2 ; out = z*dinv^2 + b2 (self-loop + bias) ----------------

__global__ void k_gemm2(const float* __restrict__ h1, const float* __restrict__ W2,
                        const float* __restrict__ b2, const float* __restrict__ dinv,
                        float* __restrict__ z, float* __restrict__ out, int n_nodes) {
    int i = blockIdx.x * blockDim.x + threadIdx.x;
    if (i >= n_nodes) return;
    const float* hr = h1 + (size_t)i * NHID;
    float z0 = 0.0f, z1 = 0.0f;
    #pragma unroll
    for (int f = 0; f < NHID; f += 4) {
        float4 hv = *(const float4*)(hr + f);
        z0 += hv.x * W2[2 * (f + 0) + 0]; z1 += hv.x * W2[2 * (f + 0) + 1];
        z0 += hv.y * W2[2 * (f + 1) + 0]; z1 += hv.y * W2[2 * (f + 1) + 1];
        z0 += hv.z * W2[2 * (f + 2) + 0]; z1 += hv.z * W2[2 * (f + 2) + 1];
        z0 += hv.w * W2[2 * (f + 3) + 0]; z1 += hv.w * W2[2 * (f + 3) + 1];
    }
    float di = dinv[i], n2 = di * di;
    z[2 * i + 0] = z0;
    z[2 * i + 1] = z1;
    out[2 * i + 0] = z0 * n2 + b2[0];
    out[2 * i + 1] = z1 * n2 + b2[1];
}

// out[dst][c] += z[src][c] * dinv[src]*dinv[dst]
__global__ void k_edge_agg2(const int* __restrict__ src, const int* __restrict__ dst,
                            const float* __restrict__ dinv, const float* __restrict__ z,
                            float* __restrict__ out, int n_edges) {
    int e = blockIdx.x * blockDim.x + threadIdx.x;
    if (e >= n_edges) return;
    int s = src[e], d = dst[e];
    float norm = dinv[s] * dinv[d];
    float2 zv = *(const float2*)(z + 2 * s);
    atomicAdd(&out[2 * d + 0], zv.x * norm);
    atomicAdd(&out[2 * d + 1], zv.y * norm);
}

// ---------------- launcher ----------------

extern "C" void kernel_launch(void* const* d_in, const int* in_sizes, int n_in,
                              void* d_out, int out_size, void* d_ws, size_t ws_size,
                              hipStream_t stream) {
    const float* x  = (const float*)d_in[0];
    const int*   ei = (const int*)d_in[1];
    const float* W1 = (const float*)d_in[2];
    const float* b1 = (const float*)d_in[3];
    const float* W2 = (const float*)d_in[4];
    const float* b2 = (const float*)d_in[5];

    int n_nodes = in_sizes[0] / NFEAT;     // 100000
    int n_edges = in_sizes[1] / 2;         // 1600000
    const int* src = ei;
    const int* dst = ei + n_edges;

    // workspace layout (floats), 256B-aligned chunks
    float* ws = (float*)d_ws;
    size_t off = 0;
    float* dinv = ws + off; off += ((size_t)n_nodes + 63) & ~(size_t)63;
    float* h    = ws + off; off += (size_t)n_nodes * NHID;      // also reused for h1
    float* out1 = ws + off; off += (size_t)n_nodes * NHID;
    float* z    = ws + off; off += ((size_t)2 * n_nodes + 63) & ~(size_t)63;
    (void)ws_size;

    float* out = (float*)d_out;

    const int B = 256;

    // 1) degree (in-degree + self-loop) -> dinv
    k_deg_init <<<(n_nodes + B - 1) / B, B, 0, stream>>>(dinv, n_nodes);
    k_deg_count<<<(n_edges + B - 1) / B, B, 0, stream>>>(dst, dinv, n_edges);
    k_dinv     <<<(n_nodes + B - 1) / B, B, 0, stream>>>(dinv, n_nodes);

    // 2) h = x @ W1  (WMMA fp32, 16-row tiles; 100000 % 16 == 0)
    int tiles  = n_nodes / 16;
    int blocks = (tiles + 7) / 8;
    k_gemm1_wmma<<<blocks, B, 0, stream>>>(x, W1, h, n_nodes);

    // 3) layer-1 aggregation: self-loop init, edge scatter, relu+bias
    long long t1 = (long long)n_nodes * NHID;
    k_selfloop1<<<(int)((t1 + B - 1) / B), B, 0, stream>>>(h, dinv, out1, n_nodes);
    long long t2 = (long long)n_edges * NHID;
    k_edge_agg1<<<(int)((t2 + B - 1) / B), B, 0, stream>>>(src, dst, dinv, h, out1, n_edges);
    k_relu_bias<<<(int)((t1 + B - 1) / B), B, 0, stream>>>(out1, b1, h, n_nodes);

    // 4) layer 2: dense transform + self-loop + bias (writes all of d_out), then edge scatter
    k_gemm2    <<<(n_nodes + B - 1) / B, B, 0, stream>>>(h, W2, b2, dinv, z, out, n_nodes);
    k_edge_agg2<<<(n_edges + B - 1) / B, B, 0, stream>>>(src, dst, dinv, z, out, n_edges);
}